// PointerNetwork_39840116638517
// MI455X (gfx1250) — compile-verified
//
#include <hip/hip_runtime.h>

typedef float v2f __attribute__((ext_vector_type(2)));
typedef float v8f __attribute__((ext_vector_type(8)));

#define B_DIM 4
#define ND    256
#define NE    512
#define C_DIM 256

// ---------------------------------------------------------------------------
// GEMM: Y[b,m,n] = sum_k X[b,m,k] * W[n,k]   (torch Linear: y = x @ W^T)
// One wave per 16x16 output tile, V_WMMA_F32_16X16X4_F32, K stepped by 4.
// Full fp32 precision (matches the fp32 reference).
// A 16x4 layout: lanes 0-15 row M=lane, VGPR0=K0/VGPR1=K1; lanes 16-31 K2/K3.
// B 4x16 layout: lanes 0-15 col N=lane, VGPR0=K0/VGPR1=K1; lanes 16-31 K2/K3.
// D 16x16 layout: VGPR r -> row (8*(lane>=16) + r), col = lane & 15.
// ---------------------------------------------------------------------------
__global__ __launch_bounds__(128) void gemm_xwt_wmma_f32(
    const float* __restrict__ X, const float* __restrict__ W,
    float* __restrict__ Y, int rows, int tiles_per_batch)
{
  const int wave = threadIdx.x >> 5;
  const int lane = threadIdx.x & 31;
  const int tile = blockIdx.x * 4 + wave;         // wave-uniform
  if (tile >= B_DIM * tiles_per_batch) return;    // uniform exit: EXEC stays all-1s

  const int b  = tile / tiles_per_batch;
  const int r  = tile - b * tiles_per_batch;
  const int tm = r >> 4;                          // tiles_n = C/16 = 16
  const int tn = r & 15;

  const int half = lane >> 4;                     // 0: lanes 0-15, 1: lanes 16-31
  const int lr   = lane & 15;

  const float* Xrow = X + (size_t)b * rows * C_DIM + (size_t)(tm * 16 + lr) * C_DIM;
  const float* Wrow = W + (size_t)(tn * 16 + lr) * C_DIM;

  v8f acc = {};
  #pragma unroll 8
  for (int k0 = 0; k0 < C_DIM; k0 += 4) {
    const int k = k0 + 2 * half;
    v2f a  = *(const v2f*)(Xrow + k);             // A[row][k], A[row][k+1]
    v2f bf = *(const v2f*)(Wrow + k);             // B[k][col] = W[col][k], contiguous
    acc = __builtin_amdgcn_wmma_f32_16x16x4_f32(false, a, false, bf,
                                                (short)0, acc, false, false);
  }

  const int col = tn * 16 + lr;
  float* Yb = Y + (size_t)b * rows * C_DIM;
  #pragma unroll
  for (int rr = 0; rr < 8; ++rr) {
    const int row = tm * 16 + half * 8 + rr;
    Yb[(size_t)row * C_DIM + col] = acc[rr];
  }
}

// ---------------------------------------------------------------------------
// Fast tanh: prefer gfx1250 V_TANH_F32; fallback = 1 - 2/(exp2(2x*log2e)+1)
// (saturates correctly at +/-1 for large |x|).
// ---------------------------------------------------------------------------
__device__ __forceinline__ float fast_tanh(float x) {
#if __has_builtin(__builtin_amdgcn_tanhf)
  return __builtin_amdgcn_tanhf(x);
#else
  const float t = __builtin_amdgcn_exp2f(x * 2.8853900817779268f); // 2*log2(e)
  return 1.0f - 2.0f * __builtin_amdgcn_rcpf(t + 1.0f);
#endif
}

// ---------------------------------------------------------------------------
// Scoring + log_softmax. One workgroup (256 thr = 8 waves) per (b, n).
// dec row + v staged in LDS; enc rows streamed as float4 (L2-resident, 256x reuse).
// Each thread owns e = t and e = t+256, then block-wide max / sum-exp reduction.
// ---------------------------------------------------------------------------
__global__ __launch_bounds__(256) void ptr_attn_score(
    const float* __restrict__ enc, const float* __restrict__ dec,
    const float* __restrict__ v, float* __restrict__ out)
{
  __shared__ float s_dec[C_DIM];
  __shared__ float s_v[C_DIM];
  __shared__ float s_red[256];

  const int t = threadIdx.x;
  const int b = blockIdx.x / ND;
  const int n = blockIdx.x - b * ND;

  s_dec[t] = dec[(size_t)(b * ND + n) * C_DIM + t];
  s_v[t]   = v[t];
  __syncthreads();

  const float4* dv4 = (const float4*)s_dec;
  const float4* vv4 = (const float4*)s_v;

  float sc[2];
  #pragma unroll
  for (int i = 0; i < 2; ++i) {
    const int e = t + i * 256;
    const float4* er = (const float4*)(enc + (size_t)(b * NE + e) * C_DIM);
    float s = 0.0f;
    #pragma unroll 4
    for (int c4 = 0; c4 < C_DIM / 4; ++c4) {
      const float4 ev = er[c4];
      const float4 dv = dv4[c4];
      const float4 vv = vv4[c4];
      s = fmaf(vv.x, fast_tanh(ev.x + dv.x), s);
      s = fmaf(vv.y, fast_tanh(ev.y + dv.y), s);
      s = fmaf(vv.z, fast_tanh(ev.z + dv.z), s);
      s = fmaf(vv.w, fast_tanh(ev.w + dv.w), s);
    }
    sc[i] = s;
  }

  // block max
  s_red[t] = fmaxf(sc[0], sc[1]);
  __syncthreads();
  #pragma unroll
  for (int off = 128; off > 0; off >>= 1) {
    if (t < off) s_red[t] = fmaxf(s_red[t], s_red[t + off]);
    __syncthreads();
  }
  const float M = s_red[0];
  __syncthreads();

  // block sum of exp
  const float L2E = 1.4426950408889634f;
  const float e0 = __builtin_amdgcn_exp2f((sc[0] - M) * L2E);
  const float e1 = __builtin_amdgcn_exp2f((sc[1] - M) * L2E);
  s_red[t] = e0 + e1;
  __syncthreads();
  #pragma unroll
  for (int off = 128; off > 0; off >>= 1) {
    if (t < off) s_red[t] += s_red[t + off];
    __syncthreads();
  }
  // ln(S) = log2(S) * ln(2)
  const float lse = M + __builtin_amdgcn_logf(s_red[0]) * 0.6931471805599453f;

  float* o = out + (size_t)blockIdx.x * NE;
  o[t]       = sc[0] - lse;
  o[t + 256] = sc[1] - lse;
}

// ---------------------------------------------------------------------------
// inputs (setup_inputs order): x_decoder(4,256,256) x_encoder(4,512,256)
//                              w1(256,256) w2(256,256) v(1,256)
// output: (4,256,512) float32 log_softmax
// workspace: enc_t 2 MiB @ +0, dec_t 1 MiB @ +2 MiB
// ---------------------------------------------------------------------------
extern "C" void kernel_launch(void* const* d_in, const int* in_sizes, int n_in,
                              void* d_out, int out_size, void* d_ws, size_t ws_size,
                              hipStream_t stream) {
  (void)in_sizes; (void)n_in; (void)out_size; (void)ws_size;
  const float* x_dec = (const float*)d_in[0];
  const float* x_enc = (const float*)d_in[1];
  const float* w1    = (const float*)d_in[2];
  const float* w2    = (const float*)d_in[3];
  const float* v     = (const float*)d_in[4];
  float*       out   = (float*)d_out;

  float* enc_t = (float*)d_ws;  // B*NE*C floats = 2 MiB
  float* dec_t = (float*)((char*)d_ws + (size_t)B_DIM * NE * C_DIM * sizeof(float));

  // enc: 4 batches * (512/16 * 256/16) = 2048 tiles, 4 waves/block -> 512 blocks
  gemm_xwt_wmma_f32<<<512, 128, 0, stream>>>(x_enc, w1, enc_t, NE, (NE / 16) * (C_DIM / 16));
  // dec: 4 batches * (256/16 * 256/16) = 1024 tiles -> 256 blocks
  gemm_xwt_wmma_f32<<<256, 128, 0, stream>>>(x_dec, w2, dec_t, ND, (ND / 16) * (C_DIM / 16));
  // scoring + log_softmax: one block per (b, n)
  ptr_attn_score<<<B_DIM * ND, 256, 0, stream>>>(enc_t, dec_t, v, out);
}